// Rwkv7SelfAttention_24154896073290
// MI455X (gfx1250) — compile-verified
//
#include <hip/hip_runtime.h>
#include <cstddef>
#include <cstdint>

// Problem constants (match reference)
#define BB   4
#define TT   4096
#define HHC  1024
#define CC   64     // head size
#define HH   16     // heads
#define DTT  16     // chunk length
#define NTT  256    // chunks

// LDS row strides (floats), padded for bank-conflict-free A-frag loads
#define TLD  68     // 64-wide tiles: bank = (4n + c) % 64, distinct per lane
#define SLD  20     // 16-wide scratch: bank = (20n + c) % 64, distinct per lane

typedef float v2f __attribute__((ext_vector_type(2)));
typedef float v4f __attribute__((ext_vector_type(4)));
typedef float v8f __attribute__((ext_vector_type(8)));
typedef int   v4i __attribute__((ext_vector_type(4)));

#if defined(__HIP_DEVICE_COMPILE__) && __has_builtin(__builtin_amdgcn_global_load_async_to_lds_b128)
#define HAVE_ASYNC 1
#else
#define HAVE_ASYNC 0
#endif

// D = A(16x4 f32) * B(4x16 f32) + C(16x16 f32)
__device__ __forceinline__ v8f wmma4(v2f a, v2f b, v8f c) {
  return __builtin_amdgcn_wmma_f32_16x16x4_f32(
      /*neg_a=*/false, a, /*neg_b=*/false, b,
      /*c_mod=*/(short)0, c, /*reuse_a=*/false, /*reuse_b=*/false);
}

__device__ __forceinline__ v2f ld2(const float* p) { return *(const v2f*)p; }

// 16-byte global -> LDS copy; async (ASYNCcnt-tracked) when available.
__device__ __forceinline__ void async_cp16(const float* g, float* l) {
#if HAVE_ASYNC
  __builtin_amdgcn_global_load_async_to_lds_b128((v4i*)g, (v4i*)l, 0, 0);
#else
  *(v4f*)l = *(const v4f*)g;
#endif
}

__device__ __forceinline__ void async_wait0() {
#if HAVE_ASYNC
#if __has_builtin(__builtin_amdgcn_s_wait_asynccnt)
  __builtin_amdgcn_s_wait_asynccnt(0);
#else
  asm volatile("s_wait_asynccnt 0" ::: "memory");
#endif
#endif
}

__global__ __launch_bounds__(128) void rwkv7_chunk_scan(
    const float* __restrict__ gw, const float* __restrict__ gq,
    const float* __restrict__ gk, const float* __restrict__ gv,
    const float* __restrict__ ga, const float* __restrict__ gb,
    float* __restrict__ gy)
{
  // Double-buffered tiles for async pipelining: [buf][tile][16][TLD]
  // tile order: 0=w 1=q 2=k 3=v 4=a 5=b
  __shared__ float sTile[2][6][DTT * TLD];
  __shared__ float sFw[CC];                // full-chunk decay per key dim
  __shared__ float sSt[4][DTT * TLD];      // per-wave state [16v][64k]
  __shared__ float sScr[4][5 * DTT * SLD]; // per-wave: ab, ak, qk, qb, u

  const int tid  = threadIdx.x;
  const int wv   = tid >> 5;        // wave id = value block (0..3)
  const int lane = tid & 31;
  const int n    = lane & 15;       // N / M / column index within fragment
  const int hi   = lane >> 4;       // half-wave
  const int kb   = 2 * hi;          // fragment K sub-offset

  const int bh = blockIdx.x;
  const int bb = bh / HH;
  const int hh = bh % HH;
  const int vcb = wv * 16;          // value-column base within head

  float* st  = &sSt[wv][0];
  float* scr = &sScr[wv][0];
  float* mAB = scr + 0 * DTT * SLD;
  float* mAK = scr + 1 * DTT * SLD;
  float* mQK = scr + 2 * DTT * SLD;
  float* mQB = scr + 3 * DTT * SLD;
  float* mU  = scr + 4 * DTT * SLD;

  // zero this wave's state slab
  for (int i = lane; i < DTT * TLD; i += 32) st[i] = 0.0f;

  const size_t baseBH = (size_t)bb * TT * HHC + (size_t)hh * CC;

  // per-tile issue: each of the 128 threads copies 2 x 16B of each tile
  auto issueTile = [&](const float* __restrict__ gp, float* sp, size_t tb) {
    #pragma unroll
    for (int it = 0; it < 2; ++it) {
      const int idx = tid + it * 128;          // 0..255
      const int row = idx >> 4;
      const int c4  = idx & 15;
      async_cp16(gp + tb + (size_t)row * HHC + (size_t)(c4 * 4),
                 sp + row * TLD + c4 * 4);
    }
  };
  auto issueChunk = [&](int ctn, int bufn) {
    const size_t tb = baseBH + (size_t)(ctn * DTT) * HHC;
    float* base = &sTile[bufn][0][0];
    issueTile(gw, base + 0 * DTT * TLD, tb);
    issueTile(gq, base + 1 * DTT * TLD, tb);
    issueTile(gk, base + 2 * DTT * TLD, tb);
    issueTile(gv, base + 3 * DTT * TLD, tb);
    issueTile(ga, base + 4 * DTT * TLD, tb);
    issueTile(gb, base + 5 * DTT * TLD, tb);
  };

  // preload chunk 0 into buffer 0
  issueChunk(0, 0);

  for (int ct = 0; ct < NTT; ++ct) {
    const int t0  = ct * DTT;
    const int cur = ct & 1;

    async_wait0();    // this wave's loads for buffer `cur` have landed
    __syncthreads();  // all waves: loads visible, previous chunk's reads done

    // prefetch next chunk into the idle buffer; overlaps with compute below
    if (ct + 1 < NTT) issueChunk(ct + 1, cur ^ 1);

    float* sW = &sTile[cur][0][0];
    float* sQ = &sTile[cur][1][0];   // becomes wq = q * incl
    float* sK = &sTile[cur][2][0];   // becomes kwi = k / incl
    float* sV = &sTile[cur][3][0];
    float* sA = &sTile[cur][4][0];   // becomes wa = a * non_incl
    float* sB = &sTile[cur][5][0];   // becomes bwi = b / incl

    // ---- decay prefix products, fused in-place (one thread per key dim) ----
    if (tid < CC) {
      const int c = tid;
      float incl = 1.0f;
      #pragma unroll
      for (int i = 0; i < DTT; ++i) {
        const int o = i * TLD + c;
        const float dec = __expf(-__expf(sW[o]));
        const float non = incl;          // exclusive prefix product
        incl *= dec;
        const float inv = 1.0f / incl;
        sQ[o] *= incl;                   // wq
        sA[o] *= non;                    // wa
        sK[o] *= inv;                    // kwi
        sB[o] *= inv;                    // bwi
      }
      sFw[c] = incl;                     // fw
    }
    __syncthreads();

    // ---- per-wave WMMA pipeline (EXEC all-ones from here) ----
    // ab = wa@bwi^T, ak = wa@kwi^T, qk = wq@kwi^T, qb = wq@bwi^T   (K=64)
    // abu = wa@state^T, yac = wq@state^T                           (K=64)
    v8f ab = {}, ak = {}, qk = {}, qb = {}, abu = {}, yac = {};
    #pragma unroll
    for (int k0 = 0; k0 < CC; k0 += 4) {
      const v2f fwa = ld2(sA + n * TLD + k0 + kb);
      const v2f fwq = ld2(sQ + n * TLD + k0 + kb);
      const v2f fbw = ld2(sB + n * TLD + k0 + kb);
      const v2f fkw = ld2(sK + n * TLD + k0 + kb);
      const v2f fst = ld2(st + n * TLD + k0 + kb);   // state[v=n][k] == (state^T) B-frag
      ab  = wmma4(fwa, fbw, ab);
      ak  = wmma4(fwa, fkw, ak);
      qk  = wmma4(fwq, fkw, qk);
      qb  = wmma4(fwq, fbw, qb);
      abu = wmma4(fwa, fst, abu);
      yac = wmma4(fwq, fst, yac);
    }

    // mask (MASK1 strict-lower for ab/ak, MASK2 incl-diag for qk/qb), bounce to LDS
    #pragma unroll
    for (int r = 0; r < 8; ++r) {
      const int m = r + 8 * hi;
      mAB[m * SLD + n] = (m >  n) ? ab[r] : 0.0f;
      mAK[m * SLD + n] = (m >  n) ? ak[r] : 0.0f;
      mQK[m * SLD + n] = (m >= n) ? qk[r] : 0.0f;
      mQB[m * SLD + n] = (m >= n) ? qb[r] : 0.0f;
    }

    // abu += ak @ sv_blk   (K=16)
    #pragma unroll
    for (int k0 = 0; k0 < DTT; k0 += 4) {
      const v2f fa = ld2(mAK + n * SLD + k0 + kb);
      const v2f fv = {sV[(k0 + kb) * TLD + vcb + n],
                      sV[(k0 + kb + 1) * TLD + vcb + n]};
      abu = wmma4(fa, fv, abu);
    }
    #pragma unroll
    for (int r = 0; r < 8; ++r) mU[(r + 8 * hi) * SLD + n] = abu[r];

    // ---- unit-lower triangular solve: u = (I - ab)^{-1} ab_u ----
    // each lane handles column n (both half-waves duplicate; same values)
    {
      float uu[DTT];
      #pragma unroll
      for (int i = 0; i < DTT; ++i) {
        float s = mU[i * SLD + n];
        #pragma unroll
        for (int j = 0; j < DTT; ++j)
          if (j < i) s += mAB[i * SLD + j] * uu[j];
        uu[i] = s;
      }
      #pragma unroll
      for (int i = 0; i < DTT; ++i) mU[i * SLD + n] = uu[i];
    }

    // ---- y = qk@sv + qb@u + wq@state^T (already in yac) ----
    #pragma unroll
    for (int k0 = 0; k0 < DTT; k0 += 4) {
      const v2f fq = ld2(mQK + n * SLD + k0 + kb);
      const v2f fv = {sV[(k0 + kb) * TLD + vcb + n],
                      sV[(k0 + kb + 1) * TLD + vcb + n]};
      yac = wmma4(fq, fv, yac);
      const v2f fb2 = ld2(mQB + n * SLD + k0 + kb);
      const v2f fu  = {mU[(k0 + kb) * SLD + n],
                       mU[(k0 + kb + 1) * SLD + n]};
      yac = wmma4(fb2, fu, yac);
    }
    #pragma unroll
    for (int r = 0; r < 8; ++r)
      gy[baseBH + (size_t)(t0 + r + 8 * hi) * HHC + vcb + n] = yac[r];

    // ---- state update: st = st*fw + sv^T@(kwi*fw) + u^T@(bwi*fw) ----
    #pragma unroll
    for (int kt = 0; kt < 4; ++kt) {
      const int kc = kt * 16 + n;
      const float fwv = sFw[kc];
      v8f stc;
      #pragma unroll
      for (int r = 0; r < 8; ++r)
        stc[r] = st[(r + 8 * hi) * TLD + kc] * fwv;
      #pragma unroll
      for (int k0 = 0; k0 < DTT; k0 += 4) {
        const v2f fsv = {sV[(k0 + kb) * TLD + vcb + n],     // (sv^T) A-frag
                         sV[(k0 + kb + 1) * TLD + vcb + n]};
        const v2f fkw = {sK[(k0 + kb) * TLD + kc] * fwv,    // (kwi*fw) B-frag
                         sK[(k0 + kb + 1) * TLD + kc] * fwv};
        stc = wmma4(fsv, fkw, stc);
        const v2f fut = {mU[(k0 + kb) * SLD + n],           // (u^T) A-frag
                         mU[(k0 + kb + 1) * SLD + n]};
        const v2f fbw = {sB[(k0 + kb) * TLD + kc] * fwv,    // (bwi*fw) B-frag
                         sB[(k0 + kb + 1) * TLD + kc] * fwv};
        stc = wmma4(fut, fbw, stc);
      }
      #pragma unroll
      for (int r = 0; r < 8; ++r)
        st[(r + 8 * hi) * TLD + kc] = stc[r];
    }
  }
}

extern "C" void kernel_launch(void* const* d_in, const int* in_sizes, int n_in,
                              void* d_out, int out_size, void* d_ws, size_t ws_size,
                              hipStream_t stream) {
  // setup_inputs order: r, w, k, v, a, b  (q := r in the reference)
  const float* r = (const float*)d_in[0];
  const float* w = (const float*)d_in[1];
  const float* k = (const float*)d_in[2];
  const float* v = (const float*)d_in[3];
  const float* a = (const float*)d_in[4];
  const float* b = (const float*)d_in[5];
  float* y = (float*)d_out;

  dim3 grid(BB * HH);   // 64 workgroups: one per (batch, head)
  dim3 block(128);      // 4 waves: one per 16-wide value block
  hipLaunchKernelGGL(rwkv7_chunk_scan, grid, block, 0, stream,
                     w, r, k, v, a, b, y);
}